// NWKernelRegression_77713138254302
// MI455X (gfx1250) — compile-verified
//
#include <hip/hip_runtime.h>
#include <stdint.h>

// Nadaraya-Watson kernel regression for MI455X (gfx1250).
// Memory-bound streaming reduction: keys+values = 512 MiB unique data, used once.
// HBM roofline @ 23.3 TB/s => ~23 us. WMMA is inapplicable (no data reuse); the
// CDNA5 feature that matters is the async global->LDS copy path (ASYNCcnt) which
// double-buffers the HBM stream against the v_exp_f32/v_fma pipeline.
// Single pass: scores s = -0.5*(w*(q-k))^2 <= 0 => exp(s) in (0,1], no overflow,
// and the softmax denominator contains exp(s_max) ~ 1 => no max-subtraction pass.

#define TPB 256                         // 8 wave32 per block, one block per row
#define F4S 2                           // float4 transfers per thread per tensor per tile
#define SLICE_FLOATS (4 * F4S)          // 8 floats = 32 B contiguous per thread
#define TILE_FLOATS (TPB * SLICE_FLOATS) // 2048 floats = 8 KB per tensor per tile
#define NEG_HALF_LOG2E 0.72134752044448170f

__global__ __launch_bounds__(TPB) void nw_regression_kernel(
    const float* __restrict__ queries,
    const float* __restrict__ keys,
    const float* __restrict__ values,
    const float* __restrict__ w_ptr,
    float* __restrict__ out,
    int n_train)
{
    __shared__ alignas(16) float sk[2][TILE_FLOATS]; // double-buffered key tiles
    __shared__ alignas(16) float sv[2][TILE_FLOATS]; // double-buffered value tiles
    __shared__ float red_l[TPB / 32];
    __shared__ float red_a[TPB / 32];

    const int row = blockIdx.x;
    const int tid = threadIdx.x;

    const float q = queries[row];
    const float w = w_ptr[0];
    // exp(-0.5*(w*(q-k))^2) == exp2(c * (q-k)^2),  c = -0.5*log2(e)*w^2
    const float c = -NEG_HALF_LOG2E * w * w;

    const float* krow = keys   + (size_t)row * (size_t)n_train;
    const float* vrow = values + (size_t)row * (size_t)n_train;

    const int ntiles = n_train / TILE_FLOATS;

    // This thread's 32B slice base in each LDS buffer. Low 32 bits of a generic
    // pointer into LDS == LDS byte offset (ISA 10.2 aperture rules).
    const uint32_t lk0 = (uint32_t)(uintptr_t)&sk[0][tid * SLICE_FLOATS];
    const uint32_t lk1 = (uint32_t)(uintptr_t)&sk[1][tid * SLICE_FLOATS];
    const uint32_t lv0 = (uint32_t)(uintptr_t)&sv[0][tid * SLICE_FLOATS];
    const uint32_t lv1 = (uint32_t)(uintptr_t)&sv[1][tid * SLICE_FLOATS];

    // Running global pointers for this thread's slice; advanced by TILE_FLOATS.
    const float* gk = krow + tid * SLICE_FLOATS;
    const float* gv = vrow + tid * SLICE_FLOATS;

    // Per-lane async DMA: INST_OFFSET is added to BOTH the LDS and the global
    // address (ISA 10, async pseudocode), so one base covers both b128 halves.
    auto prefetch_tile = [&](uint32_t lk, uint32_t lv,
                             const float* pk, const float* pv) {
        asm volatile("global_load_async_to_lds_b128 %0, %1, off"
                     :: "v"(lk), "v"(pk) : "memory");
        asm volatile("global_load_async_to_lds_b128 %0, %1, off offset:16"
                     :: "v"(lk), "v"(pk) : "memory");
        asm volatile("global_load_async_to_lds_b128 %0, %1, off"
                     :: "v"(lv), "v"(pv) : "memory");
        asm volatile("global_load_async_to_lds_b128 %0, %1, off offset:16"
                     :: "v"(lv), "v"(pv) : "memory");
    };

    float l   = 0.0f; // sum of exp(s)
    float acc = 0.0f; // sum of exp(s)*v

    if (ntiles > 0) {
        prefetch_tile(lk0, lv0, gk, gv);

#pragma unroll 1
        for (int t = 0; t < ntiles; ++t) {
            gk += TILE_FLOATS;
            gv += TILE_FLOATS;
            if (t + 1 < ntiles) {
                // Next tile into the other buffer while we consume this one.
                prefetch_tile((t & 1) ? lk0 : lk1, (t & 1) ? lv0 : lv1, gk, gv);
                // Async loads complete in order; 4 instructions per tile.
                // <=4 outstanding  ==>  tile t has fully landed in LDS.
                asm volatile("s_wait_asynccnt 0x4" ::: "memory");
            } else {
                asm volatile("s_wait_asynccnt 0x0" ::: "memory");
            }

            const int b = t & 1;
            const float4* kp = (const float4*)&sk[b][tid * SLICE_FLOATS];
            const float4* vp = (const float4*)&sv[b][tid * SLICE_FLOATS];

#pragma unroll
            for (int u = 0; u < F4S; ++u) {
                const float4 k4 = kp[u];
                const float4 v4 = vp[u];
                const float d0 = q - k4.x;
                const float d1 = q - k4.y;
                const float d2 = q - k4.z;
                const float d3 = q - k4.w;
                const float e0 = __builtin_amdgcn_exp2f(c * d0 * d0);
                const float e1 = __builtin_amdgcn_exp2f(c * d1 * d1);
                const float e2 = __builtin_amdgcn_exp2f(c * d2 * d2);
                const float e3 = __builtin_amdgcn_exp2f(c * d3 * d3);
                l += (e0 + e1) + (e2 + e3);
                acc = fmaf(e0, v4.x, acc);
                acc = fmaf(e1, v4.y, acc);
                acc = fmaf(e2, v4.z, acc);
                acc = fmaf(e3, v4.w, acc);
            }
        }
    }

    // Scalar tail (n_train not a multiple of TILE_FLOATS; no-op for 8192).
    for (int j = ntiles * TILE_FLOATS + tid; j < n_train; j += TPB) {
        const float d = q - krow[j];
        const float e = __builtin_amdgcn_exp2f(c * d * d);
        l += e;
        acc = fmaf(e, vrow[j], acc);
    }

    // Wave32 butterfly reduction.
#pragma unroll
    for (int off = 16; off >= 1; off >>= 1) {
        l   += __shfl_xor(l,   off, 32);
        acc += __shfl_xor(acc, off, 32);
    }
    const int lane = tid & 31;
    const int wave = tid >> 5;
    if (lane == 0) { red_l[wave] = l; red_a[wave] = acc; }
    __syncthreads();

    if (tid == 0) {
        float L = 0.0f, A = 0.0f;
#pragma unroll
        for (int i = 0; i < TPB / 32; ++i) { L += red_l[i]; A += red_a[i]; }
        out[row] = A / L;
    }
}

extern "C" void kernel_launch(void* const* d_in, const int* in_sizes, int n_in,
                              void* d_out, int out_size, void* d_ws, size_t ws_size,
                              hipStream_t stream) {
    const float* queries = (const float*)d_in[0];
    const float* keys    = (const float*)d_in[1];
    const float* values  = (const float*)d_in[2];
    const float* w       = (const float*)d_in[3];
    float* out = (float*)d_out;

    const int n_test  = in_sizes[0];
    const int n_train = (n_test > 0) ? (in_sizes[1] / n_test) : 0;

    nw_regression_kernel<<<n_test, TPB, 0, stream>>>(queries, keys, values, w,
                                                     out, n_train);
}